// GCN_39814346834505
// MI455X (gfx1250) — compile-verified
//
#include <hip/hip_runtime.h>

// ---------------- problem constants (match reference) ----------------
#define NNODES 100000
#define NEDGES 1600000
#define DK     128     // inner dim for all three layers (DIN = DH = 128)
#define DOUT2  64

typedef __attribute__((ext_vector_type(2))) float v2f;
typedef __attribute__((ext_vector_type(8))) float v8f;

// ---------------- degree / norm kernels ----------------
__global__ void gcn_zero2(float* __restrict__ a, float* __restrict__ b, int n) {
  int i = blockIdx.x * blockDim.x + threadIdx.x;
  if (i < n) { a[i] = 0.0f; b[i] = 0.0f; }
}

__global__ void gcn_degrees(const int* __restrict__ src, const int* __restrict__ dst,
                            float* __restrict__ deg_out, float* __restrict__ deg_in, int e) {
  int i = blockIdx.x * blockDim.x + threadIdx.x;
  if (i < e) {
    atomicAdd(&deg_out[src[i]], 1.0f);
    atomicAdd(&deg_in[dst[i]], 1.0f);
  }
}

__global__ void gcn_norms(float* __restrict__ deg_out, float* __restrict__ deg_in, int n) {
  int i = blockIdx.x * blockDim.x + threadIdx.x;
  if (i < n) {
    deg_out[i] = rsqrtf(fmaxf(deg_out[i], 1.0f));   // norm_out
    deg_in[i]  = rsqrtf(fmaxf(deg_in[i],  1.0f));   // norm_in
  }
}

// ---------------- per-layer kernels ----------------
// zero the aggregation buffer [N, DK]
__global__ void gcn_zero_agg(float4* __restrict__ agg, int n4) {
  int i = blockIdx.x * blockDim.x + threadIdx.x;
  if (i < n4) agg[i] = make_float4(0.f, 0.f, 0.f, 0.f);
}

// one wave (32 lanes) per edge; each lane moves a float4 (32*4 = 128 = DK).
// Folds norm_out[src] scaling into the read, so no staging buffer is needed:
//   agg[dst] += x[src] * norm_out[src]
__global__ void gcn_scatter(const int* __restrict__ src, const int* __restrict__ dst,
                            const float* __restrict__ x, const float* __restrict__ norm_out,
                            float* __restrict__ agg, int e) {
  int lane = threadIdx.x & 31;
  int edge = (blockIdx.x * blockDim.x + threadIdx.x) >> 5;
  if (edge >= e) return;
  int s = src[edge];
  int t = dst[edge];
  float sc = norm_out[s];
  float4 v = ((const float4*)(x + (size_t)s * DK))[lane];
  float* p = agg + (size_t)t * DK + lane * 4;
  atomicAdd(p + 0, v.x * sc);
  atomicAdd(p + 1, v.y * sc);
  atomicAdd(p + 2, v.z * sc);
  atomicAdd(p + 3, v.w * sc);
}

// out[M,DO] = relu?((agg[M,DK] * norm_in[M]) @ W[DK,DO] + bias[DO])
// One 16x16 output tile per wave via V_WMMA_F32_16X16X4_F32 (f32-exact path,
// GEMM is not the bottleneck so we keep reference precision).
//
// ISA VGPR layouts (wave32):
//  A 16x4 f32 : lane M = lane&15 ; half (lane>>4) picks K base {0,2};
//               VGPR0 = K=kb, VGPR1 = K=kb+1  -> contiguous float2 load.
//  B 4x16 f32 : lane N = lane&15 ; half picks K base {0,2} (mirror of A).
//  C/D 16x16  : 8 VGPRs; N = lane&15 ; M = v + 8*(lane>>4).
template <int DO, bool RELU>
__global__ __launch_bounds__(32 * (DO / 16))
void gcn_gemm(const float* __restrict__ agg, const float* __restrict__ norm_in,
              const float* __restrict__ W, const float* __restrict__ bias,
              float* __restrict__ out) {
  const int wave  = threadIdx.x >> 5;           // N-tile within the row block
  const int lane  = threadIdx.x & 31;
  const int tileM = blockIdx.x * 16;            // N nodes = 100000 = 6250 * 16 (exact)
  const int tileN = wave * 16;

  const int row = lane & 15;                    // A: M index
  const int col = lane & 15;                    // B/C/D: N index
  const int kb  = (lane >> 4) * 2;              // K sub-base per lane-half

  const float nrm = norm_in[tileM + row];
  const float* __restrict__ arow = agg + (size_t)(tileM + row) * DK;

  v8f c = {};
#pragma unroll
  for (int k = 0; k < DK; k += 4) {
    v2f a;
    {
      float2 av = *(const float2*)(arow + k + kb);  // K = k+kb, k+kb+1
      a.x = av.x * nrm;
      a.y = av.y * nrm;
    }
    v2f b;
    b.x = W[(size_t)(k + kb)     * DO + tileN + col];
    b.y = W[(size_t)(k + kb + 1) * DO + tileN + col];
    // 8 args: (neg_a, A, neg_b, B, c_mod, C, reuse_a, reuse_b)
    c = __builtin_amdgcn_wmma_f32_16x16x4_f32(false, a, false, b, (short)0, c,
                                              false, false);
  }

  const float bv = bias[tileN + col];
  float* __restrict__ obase = out + (size_t)tileM * DO + tileN + col;
#pragma unroll
  for (int v = 0; v < 8; ++v) {
    int m = v + 8 * (lane >> 4);
    float r = c[v] + bv;
    if (RELU) r = fmaxf(r, 0.0f);
    obase[(size_t)m * DO] = r;
  }
}

// ---------------- launcher ----------------
extern "C" void kernel_launch(void* const* d_in, const int* in_sizes, int n_in,
                              void* d_out, int out_size, void* d_ws, size_t ws_size,
                              hipStream_t stream) {
  const float* feats = (const float*)d_in[0];
  const int*   src   = (const int*)d_in[1];
  const int*   dst   = (const int*)d_in[2];
  const float* W0    = (const float*)d_in[3];
  const float* b0    = (const float*)d_in[4];
  const float* W1    = (const float*)d_in[5];
  const float* b1    = (const float*)d_in[6];
  const float* W2    = (const float*)d_in[7];
  const float* b2    = (const float*)d_in[8];

  // workspace: norm_out[N] | norm_in[N] | agg[N*DK]   (~52 MB, L2-resident)
  float* ws       = (float*)d_ws;
  float* norm_out = ws;
  float* norm_in  = ws + NNODES;
  float* agg      = ws + 2 * (size_t)NNODES;

  // output: feats | h1 | h2 | h   (concatenated flat, reference return order)
  float* out    = (float*)d_out;
  float* o_h1   = out + (size_t)NNODES * DK;
  float* o_h2   = o_h1 + (size_t)NNODES * DK;
  float* o_h    = o_h2 + (size_t)NNODES * DK;

  // feats passthrough (tuple element 0)
  hipMemcpyAsync(out, feats, (size_t)NNODES * DK * sizeof(float),
                 hipMemcpyDeviceToDevice, stream);

  // degree norms
  gcn_zero2<<<(NNODES + 255) / 256, 256, 0, stream>>>(norm_out, norm_in, NNODES);
  gcn_degrees<<<(NEDGES + 255) / 256, 256, 0, stream>>>(src, dst, norm_out, norm_in, NEDGES);
  gcn_norms<<<(NNODES + 255) / 256, 256, 0, stream>>>(norm_out, norm_in, NNODES);

  const int n4       = NNODES * (DK / 4);
  const int zgrid    = (n4 + 255) / 256;
  const int sgrid    = (NEDGES * 32 + 255) / 256;   // one wave per edge
  const int mgrid    = NNODES / 16;                 // exact: 6250 tiles

  // ---- layer 1: feats -> h1 (relu) ----
  gcn_zero_agg<<<zgrid, 256, 0, stream>>>((float4*)agg, n4);
  gcn_scatter<<<sgrid, 256, 0, stream>>>(src, dst, feats, norm_out, agg, NEDGES);
  gcn_gemm<DK, true><<<mgrid, 32 * (DK / 16), 0, stream>>>(agg, norm_in, W0, b0, o_h1);

  // ---- layer 2: h1 -> h2 (relu) ----
  gcn_zero_agg<<<zgrid, 256, 0, stream>>>((float4*)agg, n4);
  gcn_scatter<<<sgrid, 256, 0, stream>>>(src, dst, o_h1, norm_out, agg, NEDGES);
  gcn_gemm<DK, true><<<mgrid, 32 * (DK / 16), 0, stream>>>(agg, norm_in, W1, b1, o_h2);

  // ---- layer 3: h2 -> h (no act, DOUT=64) ----
  gcn_zero_agg<<<zgrid, 256, 0, stream>>>((float4*)agg, n4);
  gcn_scatter<<<sgrid, 256, 0, stream>>>(src, dst, o_h2, norm_out, agg, NEDGES);
  gcn_gemm<DOUT2, false><<<mgrid, 32 * (DOUT2 / 16), 0, stream>>>(agg, norm_in, W2, b2, o_h);
}